// GraphAttentionPooling_64424509440376
// MI455X (gfx1250) — compile-verified
//
#include <hip/hip_runtime.h>
#include <hip/hip_bf16.h>

typedef __attribute__((ext_vector_type(2))) float v2f;
typedef __attribute__((ext_vector_type(8))) float v8f;
typedef __attribute__((ext_vector_type(4))) float f4;
typedef __attribute__((ext_vector_type(4))) int   v4i;

#define F        256
#define P        3
#define GPB      16               // groups per block
#define ROWS     (GPB * P)        // 48 rows of X per block
#define STRIDE   260              // padded LDS row stride (floats): 16B aligned, %64==4 -> bank spread
#define NTHREADS 256

#if defined(__has_builtin)
# if __has_builtin(__builtin_amdgcn_global_load_async_to_lds_b128)
#  define HAVE_ASYNC_LDS 1
# endif
#endif
#ifndef HAVE_ASYNC_LDS
# define HAVE_ASYNC_LDS 0
#endif

__global__ __launch_bounds__(NTHREADS)
void GraphAttentionPooling_64424509440376_kernel(const float* __restrict__ X,
                                                 const float* __restrict__ W,
                                                 float* __restrict__ out)
{
    __shared__ float xs[ROWS * STRIDE];   // 48 KiB + pad: staged X tile
    __shared__ float ws[F];               // weight vector
    __shared__ float ss[ROWS];            // per-row scores

    const int t    = threadIdx.x;
    const int lane = t & 31;
    const int wave = t >> 5;
    const long rowBase = (long)blockIdx.x * ROWS;

    // ---------- Phase A: stage 48 rows global -> LDS (async global->LDS path) ----------
    #pragma unroll
    for (int i = 0; i < (ROWS * F / 4) / NTHREADS; ++i) {   // 12 x b128 per thread
        int idx = t + i * NTHREADS;
        int r   = idx >> 6;              // 0..47
        int c   = (idx & 63) << 2;       // 0..252 step 4 floats
        const float* g = X + (rowBase + r) * (long)F + c;
        float* l = &xs[r * STRIDE + c];
#if HAVE_ASYNC_LDS
        __builtin_amdgcn_global_load_async_to_lds_b128(
            (__attribute__((address_space(1))) v4i*)g,
            (__attribute__((address_space(3))) v4i*)l, 0, 0);
#else
        *(f4*)l = *(const f4*)g;
#endif
    }
    if (t < F / 4) {
        *(f4*)&ws[t * 4] = *(const f4*)&W[t * 4];
    }
#if HAVE_ASYNC_LDS
# if __has_builtin(__builtin_amdgcn_s_wait_asynccnt)
    __builtin_amdgcn_s_wait_asynccnt(0);
# else
    asm volatile("s_wait_asynccnt 0" ::: "memory");
# endif
#endif
    __syncthreads();

    // ---------- Phase B: waves 0..2 compute 16 row-scores each via v_wmma_f32_16x16x4_f32 ----------
    // A layout (ISA 7.12.2): lanes 0-15 hold M=lane, K={0,1} in v0,v1; lanes 16-31 hold M=lane-16, K={2,3}.
    // B holds w broadcast over all 16 columns (same K placement), so every column of D = scores.
    if (wave < P) {
        const int h    = lane >> 4;          // lane half
        const int n    = lane & 15;
        const int rloc = wave * 16 + n;      // A-matrix row for this lane
        v8f c = {};
        #pragma unroll 8
        for (int kc = 0; kc < F / 4; ++kc) {
            int k  = kc * 4 + h * 2;
            v2f a = *(const v2f*)&xs[rloc * STRIDE + k];
            v2f b = *(const v2f*)&ws[k];
            c = __builtin_amdgcn_wmma_f32_16x16x4_f32(false, a, false, b,
                                                      (short)0, c, false, false);
        }
        // D layout: VGPR v holds M = v + 8*h (all columns identical); column 0 lanes write out.
        if (n == 0) {
            #pragma unroll
            for (int v = 0; v < 8; ++v)
                ss[wave * 16 + h * 8 + v] = c[v];
        }
    }
    __syncthreads();

    // ---------- Phase C: each wave pools 2 groups (softmax over P=3, weighted row sum) ----------
    #pragma unroll
    for (int i = 0; i < GPB / 8; ++i) {
        int lg = wave * 2 + i;               // local group 0..15
        float s0 = ss[lg * 3 + 0];
        float s1 = ss[lg * 3 + 1];
        float s2 = ss[lg * 3 + 2];
        // bias is constant across the softmax axis -> exactly cancels; skip it.
        float m  = fmaxf(s0, fmaxf(s1, s2));
        float e0 = expf(s0 - m), e1 = expf(s1 - m), e2 = expf(s2 - m);
        float inv = 1.0f / (e0 + e1 + e2);
        float a0 = e0 * inv, a1 = e1 * inv, a2 = e2 * inv;

        const float* r0 = &xs[(lg * 3 + 0) * STRIDE];
        const float* r1 = &xs[(lg * 3 + 1) * STRIDE];
        const float* r2 = &xs[(lg * 3 + 2) * STRIDE];
        long gbase = ((long)blockIdx.x * GPB + lg) * (long)F;

        #pragma unroll
        for (int hh = 0; hh < 2; ++hh) {
            int c0 = hh * 128 + lane * 4;
            f4 x0 = *(const f4*)&r0[c0];
            f4 x1 = *(const f4*)&r1[c0];
            f4 x2 = *(const f4*)&r2[c0];
            f4 o;
            o.x = a0 * x0.x + a1 * x1.x + a2 * x2.x;
            o.y = a0 * x0.y + a1 * x1.y + a2 * x2.y;
            o.z = a0 * x0.z + a1 * x1.z + a2 * x2.z;
            o.w = a0 * x0.w + a1 * x1.w + a2 * x2.w;
            *(f4*)&out[gbase + c0] = o;
        }
    }
}

extern "C" void kernel_launch(void* const* d_in, const int* in_sizes, int n_in,
                              void* d_out, int out_size, void* d_ws, size_t ws_size,
                              hipStream_t stream) {
    (void)n_in; (void)out_size; (void)d_ws; (void)ws_size;
    const float* X = (const float*)d_in[0];   // [B, 256] f32
    const float* W = (const float*)d_in[1];   // [1, 256] f32 (bias d_in[2] cancels in softmax)
    float* out = (float*)d_out;               // [G, 256, 1] f32

    int B      = in_sizes[0] / F;             // 196608
    int groups = B / P;                       // 65536
    int blocks = groups / GPB;                // 4096

    GraphAttentionPooling_64424509440376_kernel<<<blocks, NTHREADS, 0, stream>>>(X, W, out);
}